// PositionalEmbedding_86646670229777
// MI455X (gfx1250) — compile-verified
//
#include <hip/hip_runtime.h>

// out[b,t,:] = tok_weight[x[b,t],:] + pos_weight[t,:]
// Pure bandwidth problem (~75 MB moved, ~3.2 us at 23.3 TB/s). No matrix op
// applies; optimize the memory path: b128 coalesced loads, scalar loads for
// block-uniform token ids, pos row reused across batches, NT stores for the
// write-once output so it does not evict the L2-resident embedding table.

typedef float v4f __attribute__((ext_vector_type(4)));

static constexpr int EMBED   = 512;
static constexpr int THREADS = EMBED / 4;  // 128 threads = 4 wave32s, 1 row/block

template <int BN>
__global__ __launch_bounds__(THREADS)
void embed_add_kernel(const int*   __restrict__ x,      // [BN, T]
                      const float* __restrict__ tok,    // [VOCAB, EMBED]
                      const float* __restrict__ pos,    // [T, EMBED]
                      float*       __restrict__ out,    // [BN, T, EMBED]
                      int T) {
    const int t   = blockIdx.x;          // one time step per block
    const int off = threadIdx.x * 4;     // 16B per thread, coalesced per wave

    // Token ids are uniform across the block -> scalar (SMEM) loads.
    int ids[BN];
#pragma unroll
    for (int b = 0; b < BN; ++b)
        ids[b] = x[(size_t)b * T + t];

    // Issue all vector loads back-to-back so 5 b128 loads are in flight
    // before the first wait (latency hiding on the HBM path).
    const v4f p = *(const v4f*)(pos + (size_t)t * EMBED + off);

    v4f w[BN];
#pragma unroll
    for (int b = 0; b < BN; ++b)
        w[b] = *(const v4f*)(tok + (size_t)ids[b] * EMBED + off);

#pragma unroll
    for (int b = 0; b < BN; ++b) {
        v4f r = w[b] + p;  // packed f32 adds
        // Write-once output: non-temporal store, keep L2 for the weight table.
        __builtin_nontemporal_store(
            r, (v4f*)(out + ((size_t)b * T + t) * EMBED + off));
    }
}

// Fallback for batch sizes other than 4 (same structure, runtime loop).
__global__ __launch_bounds__(THREADS)
void embed_add_kernel_gen(const int*   __restrict__ x,
                          const float* __restrict__ tok,
                          const float* __restrict__ pos,
                          float*       __restrict__ out,
                          int B, int T) {
    const int t   = blockIdx.x;
    const int off = threadIdx.x * 4;

    const v4f p = *(const v4f*)(pos + (size_t)t * EMBED + off);

    for (int b = 0; b < B; ++b) {
        const int id = x[(size_t)b * T + t];
        v4f w = *(const v4f*)(tok + (size_t)id * EMBED + off);
        v4f r = w + p;
        __builtin_nontemporal_store(
            r, (v4f*)(out + ((size_t)b * T + t) * EMBED + off));
    }
}

extern "C" void kernel_launch(void* const* d_in, const int* in_sizes, int n_in,
                              void* d_out, int out_size, void* d_ws, size_t ws_size,
                              hipStream_t stream) {
    (void)n_in; (void)out_size; (void)d_ws; (void)ws_size;

    const int*   x   = (const int*)  d_in[0];   // [B, T] int32 token ids
    const float* tok = (const float*)d_in[1];   // [VOCAB, EMBED] f32
    const float* pos = (const float*)d_in[2];   // [TIME, EMBED] f32

    float* out = (float*)d_out;                 // [B, T, EMBED] f32

    // Recover shapes: TIME = pos rows; reference uses T == TIME, B = |x| / T.
    const int T = in_sizes[2] / EMBED;          // 4096
    const int B = in_sizes[0] / T;              // 4

    dim3 grid(T), block(THREADS);
    if (B == 4) {
        embed_add_kernel<4><<<grid, block, 0, stream>>>(x, tok, pos, out, T);
    } else {
        embed_add_kernel_gen<<<grid, block, 0, stream>>>(x, tok, pos, out, B, T);
    }
}